// SNN_60516089201016
// MI455X (gfx1250) — compile-verified
//
#include <hip/hip_runtime.h>
#include <cstdint>

// SNN 50-step spike encoding: out[b,0,f,t], TIME innermost.
// Store-bandwidth-bound (~210 MB out vs 4 MiB in; HBM floor ~9 us). Strategy:
//   Phase 1: 1 thread = 1 element; run the 50-step recurrence in registers,
//            write spikes to an LDS tile laid out exactly like the block's
//            contiguous global output slice (stride-50-word rows are
//            bank-balanced across the 32 lanes / 64 banks).
//   Phase 2: stream the 51,200 B tile to global with CDNA5 async
//            LDS->memory stores (global_store_async_from_lds_b128),
//            fully coalesced 512 B per wave per instruction, ASYNCcnt-tracked,
//            with th:TH_STORE_NT since the 210 MB output exceeds the 192 MB L2
//            and is never re-read (avoid flushing L2 with dead write data).

#define TSTEPS 50
#define BLOCK  256

__global__ __launch_bounds__(BLOCK) void snn_encode_kernel(
    const float* __restrict__ x, float* __restrict__ out, int n) {
  __shared__ __align__(16) float tile[BLOCK * TSTEPS];  // 51,200 bytes

  const int tid = threadIdx.x;
  const long long base = (long long)blockIdx.x * BLOCK;
  const long long e = base + tid;

  // ---- Phase 1: integrate-fire-reset with adaptive threshold ----
  {
    const float c = (e < (long long)n) ? (x[e] * 1.0f + 1.0f) : 0.0f;
    float v = 0.0f;
    float th = 0.5f;
    float* row = &tile[tid * TSTEPS];
#pragma unroll
    for (int t = 0; t < TSTEPS; ++t) {
      v += c;                       // integrate
      const bool s = (v >= th);     // fire
      row[t] = s ? 1.0f : 0.0f;
      v = s ? 0.0f : v;             // reset (leak == 1.0)
      th = s ? (th + 0.03f) : th;   // threshold adaptation
    }
  }
  __syncthreads();

  // ---- Phase 2: async LDS -> global streaming stores (non-temporal) ----
  long long rem = (long long)n - base;
  const int valid = rem >= BLOCK ? BLOCK : (rem > 0 ? (int)rem : 0);
  if (valid > 0) {
    const int nfloats = valid * TSTEPS;
    const int nchunks = nfloats >> 2;  // 16-byte chunks
    float* gout = out + base * TSTEPS;
    // Generic LDS pointer: low 32 bits are the LDS byte offset (aperture layout).
    const unsigned ldsbase = (unsigned)(uintptr_t)(void*)tile;

    for (int i = tid; i < nchunks; i += BLOCK) {
      const unsigned loff = ldsbase + (unsigned)i * 16u;
      const unsigned long long gaddr =
          (unsigned long long)(uintptr_t)(gout + (size_t)i * 4);
      asm volatile("global_store_async_from_lds_b128 %0, %1, off th:TH_STORE_NT"
                   :: "v"(gaddr), "v"(loff)
                   : "memory");
    }
    // Tail (only if valid*50 not divisible by 4; never taken for this shape).
    for (int j = (nchunks << 2) + tid; j < nfloats; j += BLOCK) {
      const unsigned loff = ldsbase + (unsigned)j * 4u;
      const unsigned long long gaddr =
          (unsigned long long)(uintptr_t)(gout + (size_t)j);
      asm volatile("global_store_async_from_lds_b32 %0, %1, off th:TH_STORE_NT"
                   :: "v"(gaddr), "v"(loff)
                   : "memory");
    }
  }
  // Drain async LDS->memory traffic before the wave retires (LDS must stay live).
  asm volatile("s_wait_asynccnt 0" ::: "memory");
}

extern "C" void kernel_launch(void* const* d_in, const int* in_sizes, int n_in,
                              void* d_out, int out_size, void* d_ws, size_t ws_size,
                              hipStream_t stream) {
  const float* x = (const float*)d_in[0];
  float* out = (float*)d_out;
  const int n = in_sizes[0];  // 256*1*4096 = 1,048,576 elements
  const int grid = (n + BLOCK - 1) / BLOCK;
  hipLaunchKernelGGL(snn_encode_kernel, dim3(grid), dim3(BLOCK), 0, stream,
                     x, out, n);
}